// MultiHeadAttentionQuantum_78975858638956
// MI455X (gfx1250) — compile-verified
//
#include <hip/hip_runtime.h>

typedef __attribute__((ext_vector_type(2))) float v2f;
typedef __attribute__((ext_vector_type(8))) float v8f;

#define NB   32
#define NS   1024
#define NH   2
#define TROWS (NB * NS)        // 32768
#define OUT_EXP_ELEMS (TROWS * 8)  // 262144

// ---------------------------------------------------------------------------
// 8-qubit statevector circuit, one wave32 per row.
// Flat state index: bit (7-i) of idx <-> wire i. Lane = idx>>3, s = idx&7.
// Lanes hold 8 complex amplitudes each; cross-lane qubit ops use shfl_xor.
// ---------------------------------------------------------------------------
__global__ void __launch_bounds__(256) qc_kernel(const float* __restrict__ x,
                                                 const float* __restrict__ w,   // (2,8)
                                                 float* __restrict__ out)
{
    const int lane = threadIdx.x & 31;
    const int wave = threadIdx.x >> 5;
    const int row  = blockIdx.x * 8 + wave;

    // angles (uniform per wave)
    const float* xr = x + (size_t)row * 8;
    float ch[8], sh[8];
#pragma unroll
    for (int i = 0; i < 8; ++i) {
        float hh = 0.5f * xr[i];
        ch[i] = __cosf(hh);
        sh[i] = __sinf(hh);
    }

    // initial product state: amp = (-i)^popcount(idx) * prod(bit ? sin : cos)
    float sr[8], si[8];
#pragma unroll
    for (int s = 0; s < 8; ++s) {
        const int idx = (lane << 3) | s;
        float mag = 1.0f;
#pragma unroll
        for (int i = 0; i < 8; ++i) {
            const int bit = (idx >> (7 - i)) & 1;
            mag *= bit ? sh[i] : ch[i];
        }
        const int k = __popc(idx) & 3;
        sr[s] = (k == 0) ? mag : ((k == 2) ? -mag : 0.0f);
        si[s] = (k == 1) ? -mag : ((k == 3) ? mag : 0.0f);
    }

#pragma unroll
    for (int l = 0; l < 2; ++l) {
        // RX(w[l][i]) on each wire
#pragma unroll
        for (int i = 0; i < 8; ++i) {
            const float th = 0.5f * w[l * 8 + i];
            const float cc = __cosf(th);
            const float sn = __sinf(th);
            const int p = 7 - i;
            float pre[8], pim[8];
            if (p >= 3) {
                const int lm = 1 << (p - 3);
#pragma unroll
                for (int s = 0; s < 8; ++s) {
                    pre[s] = __shfl_xor(sr[s], lm, 32);
                    pim[s] = __shfl_xor(si[s], lm, 32);
                }
            } else {
                const int m = 1 << p;
#pragma unroll
                for (int s = 0; s < 8; ++s) {
                    pre[s] = sr[s ^ m];
                    pim[s] = si[s ^ m];
                }
            }
            float nr[8], ni[8];
#pragma unroll
            for (int s = 0; s < 8; ++s) {
                // new = cc*old + (-i*sn)*partner
                nr[s] = cc * sr[s] + sn * pim[s];
                ni[s] = cc * si[s] - sn * pre[s];
            }
#pragma unroll
            for (int s = 0; s < 8; ++s) { sr[s] = nr[s]; si[s] = ni[s]; }
        }
        // CNOT ring: ctrl=i, tgt=(i+1)%8
#pragma unroll
        for (int i = 0; i < 8; ++i) {
            const int pc = 7 - i;
            const int pt = 7 - ((i + 1) & 7);
            float pre[8], pim[8];
            if (pt >= 3) {
                const int lm = 1 << (pt - 3);
#pragma unroll
                for (int s = 0; s < 8; ++s) {
                    pre[s] = __shfl_xor(sr[s], lm, 32);
                    pim[s] = __shfl_xor(si[s], lm, 32);
                }
            } else {
                const int m = 1 << pt;
#pragma unroll
                for (int s = 0; s < 8; ++s) {
                    pre[s] = sr[s ^ m];
                    pim[s] = si[s ^ m];
                }
            }
#pragma unroll
            for (int s = 0; s < 8; ++s) {
                const int cb = (pc < 3) ? ((s >> pc) & 1) : ((lane >> (pc - 3)) & 1);
                sr[s] = cb ? pre[s] : sr[s];
                si[s] = cb ? pim[s] : si[s];
            }
        }
    }

    // PauliZ expvals: out[j] = sum probs * (1 - 2*bit(7-j))
    float pr[8], tot = 0.0f;
#pragma unroll
    for (int s = 0; s < 8; ++s) { pr[s] = sr[s] * sr[s] + si[s] * si[s]; tot += pr[s]; }
    float e[8];
#pragma unroll
    for (int j = 0; j < 8; ++j) {
        const int p = 7 - j;
        if (p >= 3) {
            e[j] = ((lane >> (p - 3)) & 1) ? -tot : tot;
        } else {
            float acc = 0.0f;
#pragma unroll
            for (int s = 0; s < 8; ++s)
                acc += ((s >> p) & 1) ? -pr[s] : pr[s];
            e[j] = acc;
        }
    }
#pragma unroll
    for (int off = 16; off >= 1; off >>= 1) {
#pragma unroll
        for (int j = 0; j < 8; ++j) e[j] += __shfl_xor(e[j], off, 32);
    }
    if (lane == 0) {
        float4* op = (float4*)(out + (size_t)row * 8);
        op[0] = make_float4(e[0], e[1], e[2], e[3]);
        op[1] = make_float4(e[4], e[5], e[6], e[7]);
    }
}

// ---------------------------------------------------------------------------
// Causal attention: block = (qtile of 16 rows, head h, batch b), 8 waves.
// Each wave computes score tiles t = wave + 8f via V_WMMA_F32_16X16X4_F32,
// keeps 8 fragments in registers, two-phase softmax via cross-wave LDS stats,
// writes normalized attn to HBM (dominant traffic) and accumulates ctx.
// ---------------------------------------------------------------------------
__global__ void __launch_bounds__(256) attn_kernel(const float* __restrict__ qe,
                                                   const float* __restrict__ ke,
                                                   const float* __restrict__ ve,
                                                   float* __restrict__ attn,
                                                   float* __restrict__ ctx)
{
    const int qtile = blockIdx.x;      // 0..63
    const int h     = blockIdx.y;      // 0..1
    const int b     = blockIdx.z;      // 0..31
    const int tid   = threadIdx.x;
    const int lane  = tid & 31;
    const int wv    = tid >> 5;        // 0..7
    const int half  = lane >> 4;       // 0 | 1
    const int l16   = lane & 15;
    const int qbase = qtile * 16;

    __shared__ float smax[8][16];
    __shared__ float ssum[8][16];
    __shared__ float gstat[16];
    __shared__ float cpart[8][16][4];

    // A fragment (ISA layout): lanes 0-15 rows M=0..15 K={0,1}; lanes 16-31 K={2,3}
    const int kk = half * 2;
    v2f afrag;
    {
        const float* qp = qe + ((size_t)(b * NS + qbase + l16)) * 8 + 4 * h + kk;
        afrag.x = qp[0]; afrag.y = qp[1];
    }

    v8f cf[8];
    float rmax[8];
#pragma unroll
    for (int r = 0; r < 8; ++r) rmax[r] = -3.0e38f;

#pragma unroll
    for (int f = 0; f < 8; ++f) {
        const int t = wv + 8 * f;                 // k-tile (wave-uniform)
        if (t <= qtile) {                         // uniform branch: EXEC stays full
            v2f bfrag;
            const int j = t * 16 + l16;
            const float* kp = ke + ((size_t)(b * NS + j)) * 8 + 4 * h + kk;
            bfrag.x = kp[0]; bfrag.y = kp[1];
            v8f cz = {};
            cf[f] = __builtin_amdgcn_wmma_f32_16x16x4_f32(
                false, afrag, false, bfrag, (short)0, cz, false, false);
            // scale 1/sqrt(4) + causal mask, track row max
#pragma unroll
            for (int r = 0; r < 8; ++r) {
                const int irow = qbase + r + half * 8;
                float vsc = cf[f][r] * 0.5f;
                if (j > irow) vsc += -1.0e9f;
                cf[f][r] = vsc;
                rmax[r] = fmaxf(rmax[r], vsc);
            }
        }
    }
    // row max across the 16 lanes of each half
#pragma unroll
    for (int off = 8; off >= 1; off >>= 1)
#pragma unroll
        for (int r = 0; r < 8; ++r)
            rmax[r] = fmaxf(rmax[r], __shfl_xor(rmax[r], off, 32));
    if (l16 == 0)
#pragma unroll
        for (int r = 0; r < 8; ++r) smax[wv][r + half * 8] = rmax[r];
    __syncthreads();
    if (tid < 16) {
        float m = smax[0][tid];
#pragma unroll
        for (int wq = 1; wq < 8; ++wq) m = fmaxf(m, smax[wq][tid]);
        gstat[tid] = m;
    }
    __syncthreads();
    float grmax[8];
#pragma unroll
    for (int r = 0; r < 8; ++r) grmax[r] = gstat[r + half * 8];
    __syncthreads();

    // exponentials + row sums
    float rsum[8];
#pragma unroll
    for (int r = 0; r < 8; ++r) rsum[r] = 0.0f;
#pragma unroll
    for (int f = 0; f < 8; ++f) {
        const int t = wv + 8 * f;
        if (t <= qtile) {
#pragma unroll
            for (int r = 0; r < 8; ++r) {
                const float p = __expf(cf[f][r] - grmax[r]);
                cf[f][r] = p;
                rsum[r] += p;
            }
        }
    }
#pragma unroll
    for (int off = 8; off >= 1; off >>= 1)
#pragma unroll
        for (int r = 0; r < 8; ++r)
            rsum[r] += __shfl_xor(rsum[r], off, 32);
    if (l16 == 0)
#pragma unroll
        for (int r = 0; r < 8; ++r) ssum[wv][r + half * 8] = rsum[r];
    __syncthreads();
    if (tid < 16) {
        float s = 0.0f;
#pragma unroll
        for (int wq = 0; wq < 8; ++wq) s += ssum[wq][tid];
        gstat[tid] = 1.0f / s;
    }
    __syncthreads();
    float ginv[8];
#pragma unroll
    for (int r = 0; r < 8; ++r) ginv[r] = gstat[r + half * 8];

    // write attn (bandwidth-dominant) + accumulate ctx = attn @ V
    float ca[8][4];
#pragma unroll
    for (int r = 0; r < 8; ++r)
#pragma unroll
        for (int d = 0; d < 4; ++d) ca[r][d] = 0.0f;

    float* abase = attn + (((size_t)(b * NH + h)) * NS + qbase) * NS;
#pragma unroll
    for (int f = 0; f < 8; ++f) {
        const int t = wv + 8 * f;
        const int j = t * 16 + l16;
        if (t <= qtile) {
            const float4 vj = *(const float4*)(ve + ((size_t)(b * NS + j)) * 8 + 4 * h);
#pragma unroll
            for (int r = 0; r < 8; ++r) {
                const float p = cf[f][r] * ginv[r];
                abase[(size_t)(r + half * 8) * NS + j] = p;
                ca[r][0] += p * vj.x;
                ca[r][1] += p * vj.y;
                ca[r][2] += p * vj.z;
                ca[r][3] += p * vj.w;
            }
        } else {
#pragma unroll
            for (int r = 0; r < 8; ++r)
                abase[(size_t)(r + half * 8) * NS + j] = 0.0f;
        }
    }
    // reduce ctx over the 16 columns handled by each half-wave
#pragma unroll
    for (int off = 8; off >= 1; off >>= 1)
#pragma unroll
        for (int r = 0; r < 8; ++r)
#pragma unroll
            for (int d = 0; d < 4; ++d)
                ca[r][d] += __shfl_xor(ca[r][d], off, 32);
    if (l16 == 0)
#pragma unroll
        for (int r = 0; r < 8; ++r)
#pragma unroll
            for (int d = 0; d < 4; ++d)
                cpart[wv][r + half * 8][d] = ca[r][d];
    __syncthreads();
    if (tid < 64) {
        const int row = tid >> 2, d = tid & 3;
        float acc = 0.0f;
#pragma unroll
        for (int wq = 0; wq < 8; ++wq) acc += cpart[wq][row][d];
        ctx[((size_t)(b * NS + qbase + row)) * 8 + 4 * h + d] = acc;
    }
}

// ---------------------------------------------------------------------------
extern "C" void kernel_launch(void* const* d_in, const int* in_sizes, int n_in,
                              void* d_out, int out_size, void* d_ws, size_t ws_size,
                              hipStream_t stream)
{
    const float* q  = (const float*)d_in[0];
    const float* k  = (const float*)d_in[1];
    const float* v  = (const float*)d_in[2];
    // d_in[3] = mask: causal, handled analytically
    const float* wq = (const float*)d_in[4];
    const float* wk = (const float*)d_in[5];
    const float* wv = (const float*)d_in[6];
    const float* wd = (const float*)d_in[7];

    float* ws = (float*)d_ws;          // 4 MB used
    float* qe = ws;
    float* ke = ws + OUT_EXP_ELEMS;
    float* ve = ws + 2 * OUT_EXP_ELEMS;
    float* cx = ws + 3 * OUT_EXP_ELEMS;

    float* outv = (float*)d_out;               // (B,S,8)
    float* attn = outv + OUT_EXP_ELEMS;        // (B,2,S,S)

    qc_kernel<<<TROWS / 8, 256, 0, stream>>>(q, wq, qe);
    qc_kernel<<<TROWS / 8, 256, 0, stream>>>(k, wk, ke);
    qc_kernel<<<TROWS / 8, 256, 0, stream>>>(v, wv, ve);
    attn_kernel<<<dim3(NS / 16, NH, NB), 256, 0, stream>>>(qe, ke, ve, attn, cx);
    qc_kernel<<<TROWS / 8, 256, 0, stream>>>(cx, wd, outv);
}